// CustomMinkowskiConvolution_8177617732130
// MI455X (gfx1250) — compile-verified
//
#include <hip/hip_runtime.h>

// Sparse voxel conv (Minkowski, stride 1) for MI455X / gfx1250.
// out[i, :] = sum_k mask[k,i] * F[in_map[k,i], :] @ W[k]   (fp32)
// N=100000, C_in=C_out=128, K=27.

#define C_IN   128
#define C_OUT  128
#define KVOL   27
#define TILE_M 64                // rows per block (4 row-tiles of 16 per wave)
#define LDS_STRIDE (C_IN + 4)    // 132 floats: row stride mod 64 = 4 -> spreads banks

typedef __attribute__((ext_vector_type(2))) float v2f;
typedef __attribute__((ext_vector_type(4))) float v4f;
typedef __attribute__((ext_vector_type(8))) float v8f;

// Packed-B size: [K][8 col-tiles][16 chunk-pairs][32 lanes] float4
#define BP_V4_COUNT (KVOL * 8 * 16 * 32)             // 110592 float4
#define BP_BYTES    ((size_t)BP_V4_COUNT * 16)       // 1.77 MB

// ---------------------------------------------------------------------------
// One-shot weight repack: Bp[(k*8+ct)*16+cp][lane] =
//   { W[k][8cp+2h  ][n], W[k][8cp+2h+1][n], W[k][8cp+4+2h][n], W[k][8cp+5+2h][n] }
// with h = lane>>4, n = 16*ct + (lane&15).  One b128 feeds two 16x16x4 WMMAs.
// ---------------------------------------------------------------------------
__global__ void pack_weights(const float* __restrict__ wgt, v4f* __restrict__ bp) {
    int idx = blockIdx.x * 256 + threadIdx.x;
    if (idx >= BP_V4_COUNT) return;
    const int lane = idx & 31;
    const int cp   = (idx >> 5) & 15;
    const int ct   = (idx >> 9) & 7;
    const int k    = idx >> 12;
    const int h    = lane >> 4;
    const int n    = ct * 16 + (lane & 15);
    const float* Wk = wgt + (size_t)k * C_IN * C_OUT;
    const int ci0 = 8 * cp + 2 * h;
    v4f v;
    v.x = Wk[(size_t)(ci0 + 0) * C_OUT + n];
    v.y = Wk[(size_t)(ci0 + 1) * C_OUT + n];
    v.z = Wk[(size_t)(ci0 + 4) * C_OUT + n];
    v.w = Wk[(size_t)(ci0 + 5) * C_OUT + n];
    bp[idx] = v;
}

// ---------------------------------------------------------------------------
// Main kernel: 256 threads = 8 waves; 64 voxel rows x 128 out channels.
// Wave w owns columns [16w,16w+16) with 4 accumulators (B reused 4x).
// Index/mask metadata is hoisted: valid bits live in a register mask, in_map
// indices in a small LDS table, so the per-k gather has no L2-latency
// dependent chain before its feature loads.
// ---------------------------------------------------------------------------
__global__ __launch_bounds__(256)
void sparse_conv_wmma_m64(const float* __restrict__ feat,
                          const v4f*   __restrict__ bp,
                          const int*   __restrict__ in_map,
                          const unsigned char* __restrict__ valid,
                          float* __restrict__ out,
                          int n_vox)
{
    __shared__ __align__(16) float tile[2][TILE_M * LDS_STRIDE];
    __shared__ int sidx[KVOL][TILE_M];     // per-row gather indices, all k

    const int r0   = blockIdx.x * TILE_M;
    const int t    = threadIdx.x;
    const int lane = t & 31;
    const int wave = t >> 5;

    // ---- gather-loader: thread t owns row (t&63), chunk-pairs 4*(t>>6)..+3
    const int lrowi = t & 63;
    const int cq    = t >> 6;
    const int lrow  = r0 + lrowi;
    const bool row_in = (lrow < n_vox);

    // ---- metadata preload (once): 27-bit valid mask in a register
    unsigned mvb = 0;
    if (row_in) {
        #pragma unroll
        for (int k = 0; k < KVOL; ++k)
            mvb |= (valid[(size_t)k * n_vox + lrow] ? 1u : 0u) << k;
    }
    // threads 0..63 fill the index table for their row
    if (t < TILE_M) {
        const int rr = r0 + t;
        const bool in = (rr < n_vox);
        #pragma unroll
        for (int k = 0; k < KVOL; ++k)
            sidx[k][t] = in ? in_map[(size_t)k * n_vox + rr] : 0;
    }
    __syncthreads();

    v4f R[8];                              // staged 32 floats (cps 4cq..4cq+3)

    auto gather = [&](int k) {
        const int gi  = sidx[k][lrowi];                    // short-latency DS read
        const bool ok = (mvb >> k) & 1u;                   // register test
        if (ok) {
            const v4f* src = (const v4f*)(feat + (size_t)gi * C_IN + cq * 32);
            #pragma unroll
            for (int i = 0; i < 8; ++i) R[i] = src[i];
        } else {
            #pragma unroll
            for (int i = 0; i < 8; ++i) R[i] = (v4f){0.f, 0.f, 0.f, 0.f};
        }
    };
    auto commit = [&](int buf) {
        float* dst = &tile[buf][lrowi * LDS_STRIDE];
        #pragma unroll
        for (int j = 0; j < 4; ++j) {
            const int cp = cq * 4 + j;
            v4f a = R[2 * j], b = R[2 * j + 1];
            v4f e = {a.x, a.y, b.x, b.y};  // even pairs -> region0 (lanes 0-15)
            v4f o = {a.z, a.w, b.z, b.w};  // odd pairs  -> region1 (lanes 16-31)
            *(v4f*)(dst + 4 * cp)      = e;
            *(v4f*)(dst + 64 + 4 * cp) = o;
        }
    };

    // ---- compute mapping (WMMA f32 16x16x4 layouts, wave32)
    const int half = lane >> 4;
    const int mrow = lane & 15;
    const int ncol = wave * 16 + (lane & 15);

    v8f acc0 = {0.f,0.f,0.f,0.f,0.f,0.f,0.f,0.f};
    v8f acc1 = acc0, acc2 = acc0, acc3 = acc0;

    gather(0);
    commit(0);
    __syncthreads();

    for (int k = 0; k < KVOL; ++k) {
        if (k + 1 < KVOL) gather(k + 1);    // loads in flight during WMMAs

        const float* A    = &tile[k & 1][mrow * LDS_STRIDE + half * 64];
        const v4f*   Bp_k = bp + ((size_t)(k * 8 + wave) * 16) * 32 + lane;

        #pragma unroll
        for (int cp = 0; cp < 16; ++cp) {
            v4f bv = Bp_k[cp * 32];                         // b128, 512B coalesced
            v2f b0 = {bv.x, bv.y}, b1 = {bv.z, bv.w};
            v4f a;
            a = *(const v4f*)(A + 0 * 16 * LDS_STRIDE + 4 * cp);   // ds_load_b128
            acc0 = __builtin_amdgcn_wmma_f32_16x16x4_f32(false, (v2f){a.x,a.y}, false, b0, (short)0, acc0, false, false);
            acc0 = __builtin_amdgcn_wmma_f32_16x16x4_f32(false, (v2f){a.z,a.w}, false, b1, (short)0, acc0, false, false);
            a = *(const v4f*)(A + 1 * 16 * LDS_STRIDE + 4 * cp);
            acc1 = __builtin_amdgcn_wmma_f32_16x16x4_f32(false, (v2f){a.x,a.y}, false, b0, (short)0, acc1, false, false);
            acc1 = __builtin_amdgcn_wmma_f32_16x16x4_f32(false, (v2f){a.z,a.w}, false, b1, (short)0, acc1, false, false);
            a = *(const v4f*)(A + 2 * 16 * LDS_STRIDE + 4 * cp);
            acc2 = __builtin_amdgcn_wmma_f32_16x16x4_f32(false, (v2f){a.x,a.y}, false, b0, (short)0, acc2, false, false);
            acc2 = __builtin_amdgcn_wmma_f32_16x16x4_f32(false, (v2f){a.z,a.w}, false, b1, (short)0, acc2, false, false);
            a = *(const v4f*)(A + 3 * 16 * LDS_STRIDE + 4 * cp);
            acc3 = __builtin_amdgcn_wmma_f32_16x16x4_f32(false, (v2f){a.x,a.y}, false, b0, (short)0, acc3, false, false);
            acc3 = __builtin_amdgcn_wmma_f32_16x16x4_f32(false, (v2f){a.z,a.w}, false, b1, (short)0, acc3, false, false);
        }

        if (k + 1 < KVOL) commit((k + 1) & 1);  // loadcnt wait lands here
        __syncthreads();
    }

    // ---- writeback: D layout: VGPR v -> M = v + 8*half (per 16-row tile)
    if (r0 + TILE_M <= n_vox) {
        #pragma unroll
        for (int v = 0; v < 8; ++v) {
            const int mb = v + 8 * half;
            out[(size_t)(r0 + 0  + mb) * C_OUT + ncol] = acc0[v];
            out[(size_t)(r0 + 16 + mb) * C_OUT + ncol] = acc1[v];
            out[(size_t)(r0 + 32 + mb) * C_OUT + ncol] = acc2[v];
            out[(size_t)(r0 + 48 + mb) * C_OUT + ncol] = acc3[v];
        }
    } else {
        #pragma unroll
        for (int v = 0; v < 8; ++v) {
            const int mb = v + 8 * half;
            if (r0 + 0  + mb < n_vox) out[(size_t)(r0 + 0  + mb) * C_OUT + ncol] = acc0[v];
            if (r0 + 16 + mb < n_vox) out[(size_t)(r0 + 16 + mb) * C_OUT + ncol] = acc1[v];
            if (r0 + 32 + mb < n_vox) out[(size_t)(r0 + 32 + mb) * C_OUT + ncol] = acc2[v];
            if (r0 + 48 + mb < n_vox) out[(size_t)(r0 + 48 + mb) * C_OUT + ncol] = acc3[v];
        }
    }
}

// ---------------------------------------------------------------------------
// Fallback (workspace too small): M=16, B via two b32 loads from [K,Cin,Cout].
// ---------------------------------------------------------------------------
__global__ __launch_bounds__(256)
void sparse_conv_wmma_direct(const float* __restrict__ feat,
                             const float* __restrict__ wgt,
                             const int*   __restrict__ in_map,
                             const unsigned char* __restrict__ valid,
                             float* __restrict__ out,
                             int n_vox)
{
    __shared__ __align__(16) float tile[2][16 * LDS_STRIDE];

    const int r0   = blockIdx.x * 16;
    const int t    = threadIdx.x;
    const int lane = t & 31;
    const int wave = t >> 5;

    const int lm   = t >> 4;
    const int lc4  = t & 15;
    const int lrow = r0 + lm;
    const bool row_in = (lrow < n_vox);

    auto load_tile = [&](int k, int buf) {
        v4f a = {0.f,0.f,0.f,0.f};
        v4f c = {0.f,0.f,0.f,0.f};
        if (row_in) {
            const int  gi = in_map[(size_t)k * n_vox + lrow];
            const bool ok = valid[(size_t)k * n_vox + lrow] != 0;
            if (ok) {
                const v4f* src = (const v4f*)(feat + (size_t)gi * C_IN);
                a = src[lc4];
                c = src[lc4 + 16];
            }
        }
        float* dst = &tile[buf][lm * LDS_STRIDE];
        *(v4f*)(dst + 4 * lc4)        = a;
        *(v4f*)(dst + 4 * (lc4 + 16)) = c;
    };

    const int half = lane >> 4;
    const int mrow = lane & 15;
    const int ncol = wave * 16 + (lane & 15);

    v8f acc = {0.f,0.f,0.f,0.f,0.f,0.f,0.f,0.f};

    load_tile(0, 0);
    __syncthreads();

    for (int k = 0; k < KVOL; ++k) {
        if (k + 1 < KVOL) load_tile(k + 1, (k + 1) & 1);

        const float* A  = &tile[k & 1][mrow * LDS_STRIDE + 2 * half];
        const float* Bk = wgt + (size_t)k * C_IN * C_OUT;

        #pragma unroll 4
        for (int c = 0; c < C_IN / 4; ++c) {
            v2f av = *(const v2f*)(A + 4 * c);
            const int ci = 4 * c + 2 * half;
            v2f bv;
            bv.x = Bk[(size_t)ci * C_OUT + ncol];
            bv.y = Bk[(size_t)(ci + 1) * C_OUT + ncol];
            acc = __builtin_amdgcn_wmma_f32_16x16x4_f32(
                false, av, false, bv, (short)0, acc, false, false);
        }
        __syncthreads();
    }

    #pragma unroll
    for (int v = 0; v < 8; ++v) {
        const int m = v + 8 * half;
        if (r0 + m < n_vox)
            out[(size_t)(r0 + m) * C_OUT + ncol] = acc[v];
    }
}

extern "C" void kernel_launch(void* const* d_in, const int* in_sizes, int n_in,
                              void* d_out, int out_size, void* d_ws, size_t ws_size,
                              hipStream_t stream) {
    (void)n_in; (void)out_size;
    const float* feat  = (const float*)d_in[0];                  // [N,128] fp32
    const float* wgt   = (const float*)d_in[1];                  // [27,128,128] fp32
    const int*   imap  = (const int*)d_in[2];                    // [27,N] int32
    const unsigned char* vmask = (const unsigned char*)d_in[3];  // [27,N] bool
    float* out = (float*)d_out;

    const int n_vox = in_sizes[0] / C_IN;                        // 100000

    if (d_ws && ws_size >= BP_BYTES) {
        v4f* bp = (v4f*)d_ws;
        pack_weights<<<(BP_V4_COUNT + 255) / 256, 256, 0, stream>>>(wgt, bp);
        const int blocks = (n_vox + TILE_M - 1) / TILE_M;        // 1563
        sparse_conv_wmma_m64<<<blocks, 256, 0, stream>>>(feat, bp, imap, vmask, out, n_vox);
    } else {
        const int blocks = (n_vox + 15) / 16;                    // 6250
        sparse_conv_wmma_direct<<<blocks, 256, 0, stream>>>(feat, wgt, imap, vmask, out, n_vox);
    }
}